// GraphConvLayer_47777216201170
// MI455X (gfx1250) — compile-verified
//
#include <hip/hip_runtime.h>
#include <hip/hip_bf16.h>

typedef __attribute__((ext_vector_type(2))) float v2f;
typedef __attribute__((ext_vector_type(8))) float v8f;

#define BATCH 8
#define NNODE 10000
#define DIM   128
#define NEDGE 320000
#define MROWS (BATCH * NNODE)          // 80000
#define TOTAL ((long)MROWS * DIM)      // 10,240,000
#define BN_EPS 1e-5f

// ---------------------------------------------------------------------------
// Phase 0: zero scratch regions
// ---------------------------------------------------------------------------
__global__ void zero_kernel(float* __restrict__ p, long n) {
    long i = (long)blockIdx.x * blockDim.x + threadIdx.x;
    long stride = (long)gridDim.x * blockDim.x;
    for (; i < n; i += stride) p[i] = 0.0f;
}

// ---------------------------------------------------------------------------
// Phase 1: x_t = x @ W^T + b   via V_WMMA_F32_16X16X4_F32
//   x  : (80000, 128) row-major
//   W  : (128, 128) row-major  [out_ch][in_ch]
//   xt : (80000, 128)
// Block: 256 thr = 8 waves; each wave: 16 rows x 128 cols (8 tiles of 16x16).
// Grid: 80000/128 = 625 blocks.
// ---------------------------------------------------------------------------
__global__ __launch_bounds__(256) void gemm_bias_kernel(
    const float* __restrict__ x,
    const float* __restrict__ W,
    const float* __restrict__ bias,
    float* __restrict__ xt)
{
    // W transposed into LDS: Ws[k*128 + n] = W[n*128 + k]
    __shared__ float Ws[DIM * DIM];     // 64 KB
    for (int i = threadIdx.x; i < DIM * DIM; i += 256) {
        int n = i >> 7;
        int k = i & 127;
        Ws[k * DIM + n] = W[i];
    }
    __syncthreads();

    const int wave = threadIdx.x >> 5;
    const int lane = threadIdx.x & 31;
    const long row_base = (long)blockIdx.x * 128 + wave * 16;

    const int mrow  = lane & 15;          // M index within tile (A) / N index (B,C)
    const int khalf = (lane >> 4) * 2;    // 0 or 2: which K pair this half-wave holds

    v8f acc[8];
#pragma unroll
    for (int j = 0; j < 8; ++j) acc[j] = (v8f)(0.0f);

    const float* xrow = x + (row_base + mrow) * DIM;

    for (int k0 = 0; k0 < DIM; k0 += 4) {
        // A fragment 16x4: VGPR0 = K(k0+khalf), VGPR1 = K(k0+khalf+1)
        v2f a;
        a.x = xrow[k0 + khalf];
        a.y = xrow[k0 + khalf + 1];
#pragma unroll
        for (int j = 0; j < 8; ++j) {
            // B fragment 4x16: B[k][n] = W[n][k] = Ws[k*128 + n]
            v2f bf;
            const int n = j * 16 + mrow;
            bf.x = Ws[(k0 + khalf) * DIM + n];
            bf.y = Ws[(k0 + khalf + 1) * DIM + n];
            acc[j] = __builtin_amdgcn_wmma_f32_16x16x4_f32(
                false, a, false, bf, (short)0, acc[j], false, false);
        }
    }

    // C/D layout: VGPR i, lanes 0-15 -> M=i ; lanes 16-31 -> M=i+8 ; N = lane&15
    const int ncol = lane & 15;
    const int mofs = (lane >> 4) * 8;
#pragma unroll
    for (int j = 0; j < 8; ++j) {
        const float bv = bias[j * 16 + ncol];
        float* dst = xt + (row_base + mofs) * DIM + j * 16 + ncol;
#pragma unroll
        for (int i = 0; i < 8; ++i) {
            dst[(long)i * DIM] = acc[j][i] + bv;
        }
    }
}

// ---------------------------------------------------------------------------
// Phase 2: degree over SOURCE indices (row), fp32 counts
// ---------------------------------------------------------------------------
__global__ void degree_kernel(const long long* __restrict__ ei,
                              float* __restrict__ degree) {
    int e = blockIdx.x * blockDim.x + threadIdx.x;
    if (e < NEDGE) {
        atomicAdd(&degree[(int)ei[e]], 1.0f);
    }
}

// ---------------------------------------------------------------------------
// Phase 3: scatter-add  agg[b, col[e], :] += xt[b, row[e], :]
// 32 lanes per edge, float4 per lane -> one coalesced 512B row per wave.
// ---------------------------------------------------------------------------
__global__ __launch_bounds__(256) void scatter_kernel(
    const float* __restrict__ xt,
    const long long* __restrict__ ei,
    float* __restrict__ agg)
{
    long t = (long)blockIdx.x * blockDim.x + threadIdx.x;
    if (t >= (long)NEDGE * 32) return;
    const int e  = (int)(t >> 5);
    const int c0 = (int)(t & 31) * 4;
    const int src = (int)ei[e];
    const int dst = (int)ei[NEDGE + e];
#pragma unroll
    for (int b = 0; b < BATCH; ++b) {
        const float4 v = *(const float4*)(xt + ((long)(b * NNODE + src)) * DIM + c0);
        float* a = agg + ((long)(b * NNODE + dst)) * DIM + c0;
        atomicAdd(a + 0, v.x);
        atomicAdd(a + 1, v.y);
        atomicAdd(a + 2, v.z);
        atomicAdd(a + 3, v.w);
    }
}

// ---------------------------------------------------------------------------
// Phase 4: out = agg/clip(deg,1) + xt (in place in agg) + per-channel sum/sumsq
// Launch with gridDim*blockDim a multiple of 128 so each thread owns ONE channel.
// ---------------------------------------------------------------------------
__global__ __launch_bounds__(256) void fuse_out_stats_kernel(
    const float* __restrict__ xt,
    float* __restrict__ agg,
    const float* __restrict__ degree,
    float* __restrict__ sums,
    float* __restrict__ sumsq)
{
    const long stride = (long)gridDim.x * blockDim.x;   // multiple of 128
    float lsum = 0.0f, lsq = 0.0f;
    for (long idx = (long)blockIdx.x * blockDim.x + threadIdx.x; idx < TOTAL; idx += stride) {
        const int n = (int)((idx >> 7) % NNODE);
        const float d = fmaxf(degree[n], 1.0f);
        const float v = agg[idx] / d + xt[idx];
        agg[idx] = v;
        lsum += v;
        lsq  += v * v;
    }
    __shared__ float s_sum[256];
    __shared__ float s_sq[256];
    s_sum[threadIdx.x] = lsum;
    s_sq[threadIdx.x]  = lsq;
    __syncthreads();
    if (threadIdx.x < 128) {
        // thread tid and tid+128 accumulate the same channel (tid & 127)
        atomicAdd(&sums[threadIdx.x],  s_sum[threadIdx.x] + s_sum[threadIdx.x + 128]);
        atomicAdd(&sumsq[threadIdx.x], s_sq[threadIdx.x]  + s_sq[threadIdx.x + 128]);
    }
}

// ---------------------------------------------------------------------------
// Phase 5: BN scale/shift from batch statistics
// ---------------------------------------------------------------------------
__global__ void finalize_stats_kernel(const float* __restrict__ sums,
                                      const float* __restrict__ sumsq,
                                      const float* __restrict__ gamma,
                                      const float* __restrict__ beta,
                                      float* __restrict__ scale,
                                      float* __restrict__ shift) {
    const int c = threadIdx.x;   // 128 threads
    const float inv = 1.0f / (float)MROWS;
    const float mean = sums[c] * inv;
    const float var  = sumsq[c] * inv - mean * mean;
    const float sc   = gamma[c] * rsqrtf(var + BN_EPS);
    scale[c] = sc;
    shift[c] = beta[c] - mean * sc;
}

// ---------------------------------------------------------------------------
// Phase 6: out = relu(v*scale + shift), float4 vectorized
// ---------------------------------------------------------------------------
__global__ __launch_bounds__(256) void bn_relu_kernel(
    const float4* __restrict__ v,
    const float* __restrict__ scale,
    const float* __restrict__ shift,
    float4* __restrict__ out,
    long total4)
{
    long i = (long)blockIdx.x * blockDim.x + threadIdx.x;
    if (i >= total4) return;
    const int c0 = (int)((i * 4) & 127);
    const float4 a = v[i];
    float4 r;
    r.x = fmaxf(fmaf(a.x, scale[c0 + 0], shift[c0 + 0]), 0.0f);
    r.y = fmaxf(fmaf(a.y, scale[c0 + 1], shift[c0 + 1]), 0.0f);
    r.z = fmaxf(fmaf(a.z, scale[c0 + 2], shift[c0 + 2]), 0.0f);
    r.w = fmaxf(fmaf(a.w, scale[c0 + 3], shift[c0 + 3]), 0.0f);
    out[i] = r;
}

// ---------------------------------------------------------------------------
// Launch
// ---------------------------------------------------------------------------
extern "C" void kernel_launch(void* const* d_in, const int* in_sizes, int n_in,
                              void* d_out, int out_size, void* d_ws, size_t ws_size,
                              hipStream_t stream) {
    (void)in_sizes; (void)n_in; (void)out_size; (void)ws_size;

    const float*     x     = (const float*)d_in[0];
    const long long* ei    = (const long long*)d_in[1];   // int64 (2, E)
    const float*     W     = (const float*)d_in[2];
    const float*     bias  = (const float*)d_in[3];
    const float*     gamma = (const float*)d_in[4];
    const float*     beta  = (const float*)d_in[5];
    float*           out   = (float*)d_out;

    // Workspace layout (floats)
    float* ws     = (float*)d_ws;
    float* xt     = ws;                       // 10,240,000
    float* agg    = ws + TOTAL;               // 10,240,000
    float* degree = ws + 2 * TOTAL;           // 10,000
    float* sums   = ws + 2 * TOTAL + 16000;   // 128
    float* sumsq  = sums + 128;               // 128
    float* scale  = sums + 256;               // 128
    float* shift  = sums + 384;               // 128

    // 0) zero agg+degree (contiguous) and stats
    zero_kernel<<<2048, 256, 0, stream>>>(agg, TOTAL + 16000 + 512);

    // 1) GEMM + bias (WMMA fp32)
    gemm_bias_kernel<<<MROWS / 128, 256, 0, stream>>>(x, W, bias, xt);

    // 2) degree over source indices
    degree_kernel<<<(NEDGE + 255) / 256, 256, 0, stream>>>(ei, degree);

    // 3) edge scatter-add
    {
        long nthreads = (long)NEDGE * 32;
        scatter_kernel<<<(unsigned)((nthreads + 255) / 256), 256, 0, stream>>>(xt, ei, agg);
    }

    // 4) out = agg/deg + xt (in-place in agg) + channel stats
    fuse_out_stats_kernel<<<1024, 256, 0, stream>>>(xt, agg, degree, sums, sumsq);

    // 5) BN scale/shift
    finalize_stats_kernel<<<1, 128, 0, stream>>>(sums, sumsq, gamma, beta, scale, shift);

    // 6) BN apply + ReLU -> d_out
    {
        long total4 = TOTAL / 4;
        bn_relu_kernel<<<(unsigned)((total4 + 255) / 256), 256, 0, stream>>>(
            (const float4*)agg, scale, shift, (float4*)out, total4);
    }
}